// Model_23888608100983
// MI455X (gfx1250) — compile-verified
//
#include <hip/hip_runtime.h>
#include <hip/hip_bf16.h>

// ---------------- model constants ----------------
#define NSEQ   1024          // B*C sequences
#define SEQ    63            // patches per sequence (P)
#define DMODEL 256
#define NHEAD  4
#define DHEAD  64
#define NUPD   384           // UP
#define NROWS  (NSEQ*SEQ)    // 64512 rows of (n,s)
#define PRED   96
#define HPITCH 260           // padded LDS row pitch (bank-conflict free)

typedef float v2f __attribute__((ext_vector_type(2)));
typedef float v8f __attribute__((ext_vector_type(8)));

// Full-precision CDNA5 WMMA: D(16x16 f32) = A(16x4 f32) * B(4x16 f32) + C
__device__ __forceinline__ v8f wmma4(v2f a, v2f b, v8f c) {
  return __builtin_amdgcn_wmma_f32_16x16x4_f32(
      /*neg_a=*/false, a, /*neg_b=*/false, b,
      /*c_mod=*/(short)0, c, /*reuse_a=*/false, /*reuse_b=*/false);
}

__device__ __forceinline__ float sigmoidf_(float x) {
  return 1.0f / (1.0f + __expf(-x));
}
__device__ __forceinline__ float logsigf_(float x) {
  return (x >= 0.0f) ? -log1pf(expf(-x)) : (x - log1pf(expf(x)));
}
__device__ __forceinline__ float gelu_tanh_(float x) {
  const float k0 = 0.7978845608028654f; // sqrt(2/pi)
  float x3 = x * x * x;
  return 0.5f * x * (1.0f + tanhf(k0 * (x + 0.044715f * x3)));
}

// ---------------- patch embed: h = patches @ W_emb + b_emb ----------------
__global__ void embed_kernel(const float* __restrict__ x,
                             const float* __restrict__ Wemb,
                             const float* __restrict__ bemb,
                             float* __restrict__ h) {
  int wave = (blockIdx.x * blockDim.x + threadIdx.x) >> 5;
  int lane = threadIdx.x & 31;
  int mtile = wave >> 4;
  int ntile = wave & 15;
  if (mtile >= NROWS / 16) return;
  int row = mtile * 16 + (lane & 15);
  int n = row / SEQ, p = row % SEQ;
  int b = n >> 5, c = n & 31;
  int kd = (lane >> 4) << 1;
  int ncol = ntile * 16 + (lane & 15);
  const float* xb = x + (size_t)b * 512 * 32 + c;
  v8f acc = {};
#pragma unroll
  for (int k = 0; k < 16; k += 4) {
    int kk = k + kd;
    int l0 = p * 8 + kk;
    v2f a; a.x = xb[(size_t)l0 * 32]; a.y = xb[(size_t)(l0 + 1) * 32];
    v2f bb; bb.x = Wemb[kk * DMODEL + ncol]; bb.y = Wemb[(kk + 1) * DMODEL + ncol];
    acc = wmma4(a, bb, acc);
  }
  float bias = bemb[ncol];
  int mbase = (lane >> 4) << 3;
#pragma unroll
  for (int r = 0; r < 8; ++r) {
    int orow = mtile * 16 + mbase + r;
    h[(size_t)orow * DMODEL + ncol] = acc[r] + bias;
  }
}

// ---------------- layernorm over D=256, one wave per row ----------------
__global__ void ln_kernel(const float* __restrict__ in,
                          const float* __restrict__ w,
                          const float* __restrict__ bb,
                          float* __restrict__ out, int rows) {
  int r = (blockIdx.x * blockDim.x + threadIdx.x) >> 5;
  int lane = threadIdx.x & 31;
  if (r >= rows) return;
  const float* rp = in + (size_t)r * DMODEL;
  float v[8], s = 0.0f, s2 = 0.0f;
#pragma unroll
  for (int i = 0; i < 8; ++i) {
    v[i] = rp[lane + i * 32];
    s += v[i]; s2 += v[i] * v[i];
  }
#pragma unroll
  for (int o = 16; o > 0; o >>= 1) {
    s  += __shfl_xor(s, o, 32);
    s2 += __shfl_xor(s2, o, 32);
  }
  float mu  = s * (1.0f / DMODEL);
  float var = s2 * (1.0f / DMODEL) - mu * mu;
  float inv = rsqrtf(var + 1e-5f);
  float* op = out + (size_t)r * DMODEL;
#pragma unroll
  for (int i = 0; i < 8; ++i) {
    int d = lane + i * 32;
    op[d] = (v[i] - mu) * inv * w[d] + bb[d];
  }
}

// ---------------- causal conv4 + SiLU ----------------
__global__ void conv_silu_kernel(const float* __restrict__ xln,
                                 const float* __restrict__ cw,
                                 const float* __restrict__ cb,
                                 float* __restrict__ xc) {
  size_t idx = (size_t)blockIdx.x * blockDim.x + threadIdx.x;
  if (idx >= (size_t)NROWS * DMODEL) return;
  int d = (int)(idx & (DMODEL - 1));
  size_t r = idx >> 8;
  int s = (int)(r % SEQ);
  size_t nbase = (r - (size_t)s) * DMODEL;
  float acc = cb[d];
#pragma unroll
  for (int k = 0; k < 4; ++k) {
    int ss = s - 3 + k;
    if (ss >= 0) acc += xln[nbase + (size_t)ss * DMODEL + d] * cw[k * DMODEL + d];
  }
  xc[idx] = acc * sigmoidf_(acc);
}

// ---------------- persistent sLSTM scan ----------------
// One workgroup (16 waves, 512 threads) owns 16 sequences for all 63 steps.
// Wave w owns column-tile w (cols w*16..w*16+15, head = w>>2) and computes ALL
// 4 gate tiles for those columns; gate pre-activations stay in registers, the
// exponential-gating state (c,n,m) lives in the owning lanes, and only the
// hidden state h round-trips through LDS (padded pitch, conflict-free).
// Next step's activation rows are prefetched (global_prefetch_b8) to overlap
// L2 latency with the current step's WMMA work.
__global__ void __launch_bounds__(512) scan_kernel(
    const float* __restrict__ xc, const float* __restrict__ xln,
    const float* __restrict__ Wg, const float* __restrict__ Rg,
    const float* __restrict__ gbv, float* __restrict__ y) {
  __shared__ float lds_h[16 * HPITCH];
  int tid = threadIdx.x;
  int wave = tid >> 5;
  int lane = tid & 31;
  int nbase = blockIdx.x * 16;

  int ct   = wave;                 // column tile 0..15
  int head = ct >> 2;
  int col  = ct * 16 + (lane & 15);  // global d (0..255)
  int ch   = col & 63;               // column within head
  int mrow = lane & 15;              // local sequence for A fragments
  int kd   = (lane >> 4) << 1;
  int mbase = (lane >> 4) << 3;

  const float* W0 = Wg + ((size_t)(0 * NHEAD + head)) * DHEAD * DHEAD;
  const float* W1 = Wg + ((size_t)(1 * NHEAD + head)) * DHEAD * DHEAD;
  const float* W2 = Wg + ((size_t)(2 * NHEAD + head)) * DHEAD * DHEAD;
  const float* W3 = Wg + ((size_t)(3 * NHEAD + head)) * DHEAD * DHEAD;
  const float* R0 = Rg + ((size_t)(0 * NHEAD + head)) * DHEAD * DHEAD;
  const float* R1 = Rg + ((size_t)(1 * NHEAD + head)) * DHEAD * DHEAD;
  const float* R2 = Rg + ((size_t)(2 * NHEAD + head)) * DHEAD * DHEAD;
  const float* R3 = Rg + ((size_t)(3 * NHEAD + head)) * DHEAD * DHEAD;
  float bias0 = gbv[0 * DMODEL + col];
  float bias1 = gbv[1 * DMODEL + col];
  float bias2 = gbv[2 * DMODEL + col];
  float bias3 = gbv[3 * DMODEL + col];

  float c_r[8], n_r[8], m_r[8];
#pragma unroll
  for (int r = 0; r < 8; ++r) { c_r[r] = 0.0f; n_r[r] = 0.0f; m_r[r] = 0.0f; }
  for (int i = tid; i < 16 * HPITCH; i += 512) lds_h[i] = 0.0f;
  __syncthreads();

  const float* axc_base = xc  + (size_t)(nbase + mrow) * SEQ * DMODEL + head * DHEAD;
  const float* axl_base = xln + (size_t)(nbase + mrow) * SEQ * DMODEL + head * DHEAD;
  const float* hrow = lds_h + mrow * HPITCH + head * DHEAD;

  for (int s = 0; s < SEQ; ++s) {
    const float* axc = axc_base + (size_t)s * DMODEL;
    const float* axl = axl_base + (size_t)s * DMODEL;
    // prefetch next step's activation rows (wave-uniform guard keeps EXEC full)
    if (s + 1 < SEQ) {
      __builtin_prefetch(axc + DMODEL + kd, 0, 3);
      __builtin_prefetch(axl + DMODEL + kd, 0, 3);
    }
    v8f a0 = {}, a1 = {}, a2 = {}, a3 = {};
    // input-gate contributions: gates {i,f} from conv-SiLU, {z,o} from LN
    for (int k = 0; k < DHEAD; k += 4) {
      int kk = k + kd;
      v2f ac; ac.x = axc[kk]; ac.y = axc[kk + 1];
      v2f al; al.x = axl[kk]; al.y = axl[kk + 1];
      v2f b;
      b.x = W0[kk * DHEAD + ch]; b.y = W0[(kk + 1) * DHEAD + ch]; a0 = wmma4(ac, b, a0);
      b.x = W1[kk * DHEAD + ch]; b.y = W1[(kk + 1) * DHEAD + ch]; a1 = wmma4(ac, b, a1);
      b.x = W2[kk * DHEAD + ch]; b.y = W2[(kk + 1) * DHEAD + ch]; a2 = wmma4(al, b, a2);
      b.x = W3[kk * DHEAD + ch]; b.y = W3[(kk + 1) * DHEAD + ch]; a3 = wmma4(al, b, a3);
    }
    // recurrent contributions: A from LDS hidden state, shared by all 4 gates
    for (int k = 0; k < DHEAD; k += 4) {
      int kk = k + kd;
      v2f ah; ah.x = hrow[kk]; ah.y = hrow[kk + 1];
      v2f b;
      b.x = R0[kk * DHEAD + ch]; b.y = R0[(kk + 1) * DHEAD + ch]; a0 = wmma4(ah, b, a0);
      b.x = R1[kk * DHEAD + ch]; b.y = R1[(kk + 1) * DHEAD + ch]; a1 = wmma4(ah, b, a1);
      b.x = R2[kk * DHEAD + ch]; b.y = R2[(kk + 1) * DHEAD + ch]; a2 = wmma4(ah, b, a2);
      b.x = R3[kk * DHEAD + ch]; b.y = R3[(kk + 1) * DHEAD + ch]; a3 = wmma4(ah, b, a3);
    }
    __syncthreads();   // everyone done reading old lds_h
    // exponential-gating pointwise update, entirely in registers
#pragma unroll
    for (int r = 0; r < 8; ++r) {
      float ip = a0[r] + bias0;
      float fp = a1[r] + bias1;
      float zp = a2[r] + bias2;
      float op = a3[r] + bias3;
      float logf_ = m_r[r] + logsigf_(fp);
      float mn = fmaxf(ip, logf_);
      float i = expf(ip - mn);
      float f = expf(logf_ - mn);
      float cn = f * c_r[r] + i * tanhf(zp);
      float nn2 = f * n_r[r] + i;
      float hv = sigmoidf_(op) * cn / nn2;
      c_r[r] = cn; n_r[r] = nn2; m_r[r] = mn;
      int lrow = mbase + r;
      lds_h[lrow * HPITCH + col] = hv;
      y[((size_t)(nbase + lrow) * SEQ + s) * DMODEL + col] = hv;
    }
    __syncthreads();   // new lds_h visible for next step
  }
}

// ---------------- per-head group-norm + scale + residual ----------------
__global__ void gnorm_res_kernel(const float* __restrict__ y,
                                 const float* __restrict__ gnw,
                                 float* __restrict__ h) {
  int wid = (blockIdx.x * blockDim.x + threadIdx.x) >> 5;
  int lane = threadIdx.x & 31;
  if (wid >= NROWS * NHEAD) return;
  int hh = wid & 3;
  int r = wid >> 2;
  const float* yp = y + (size_t)r * DMODEL + hh * DHEAD;
  float v0 = yp[lane], v1 = yp[lane + 32];
  float s = v0 + v1, s2 = v0 * v0 + v1 * v1;
#pragma unroll
  for (int o = 16; o > 0; o >>= 1) {
    s  += __shfl_xor(s, o, 32);
    s2 += __shfl_xor(s2, o, 32);
  }
  float mu  = s * (1.0f / DHEAD);
  float var = s2 * (1.0f / DHEAD) - mu * mu;
  float inv = rsqrtf(var + 1e-5f);
  float* hp = h + (size_t)r * DMODEL + hh * DHEAD;
  hp[lane]      += (v0 - mu) * inv * gnw[hh * DHEAD + lane];
  hp[lane + 32] += (v1 - mu) * inv * gnw[hh * DHEAD + lane + 32];
}

// ---------------- up-proj (dual tile) + gelu gating: v = gelu(g1)*g2 ----------------
__global__ void up_gelu_kernel(const float* __restrict__ a,
                               const float* __restrict__ upw,
                               const float* __restrict__ upb,
                               float* __restrict__ v) {
  int wave = (blockIdx.x * blockDim.x + threadIdx.x) >> 5;
  int lane = threadIdx.x & 31;
  int mtile = wave / 24, ntile = wave % 24;
  if (mtile >= NROWS / 16) return;
  int row = mtile * 16 + (lane & 15);
  int col = ntile * 16 + (lane & 15);
  int kd = (lane >> 4) << 1;
  const float* ap = a + (size_t)row * DMODEL;
  v8f acc1 = {}, acc2 = {};
#pragma unroll 4
  for (int k = 0; k < DMODEL; k += 4) {
    int kk = k + kd;
    v2f av; av.x = ap[kk]; av.y = ap[kk + 1];
    const float* w0 = upw + (size_t)kk * (2 * NUPD) + col;
    const float* w1 = w0 + (2 * NUPD);
    v2f b1; b1.x = w0[0];    b1.y = w1[0];
    v2f b2; b2.x = w0[NUPD]; b2.y = w1[NUPD];
    acc1 = wmma4(av, b1, acc1);
    acc2 = wmma4(av, b2, acc2);
  }
  float ub1 = upb[col], ub2 = upb[col + NUPD];
  int mbase = (lane >> 4) << 3;
#pragma unroll
  for (int r = 0; r < 8; ++r) {
    float g1 = acc1[r] + ub1, g2 = acc2[r] + ub2;
    int orow = mtile * 16 + mbase + r;
    v[(size_t)orow * NUPD + col] = gelu_tanh_(g1) * g2;
  }
}

// ---------------- down-proj + bias + residual into h ----------------
__global__ void down_res_kernel(const float* __restrict__ v,
                                const float* __restrict__ dnw,
                                const float* __restrict__ dnb,
                                float* __restrict__ h) {
  int wave = (blockIdx.x * blockDim.x + threadIdx.x) >> 5;
  int lane = threadIdx.x & 31;
  int mtile = wave >> 4, ntile = wave & 15;
  if (mtile >= NROWS / 16) return;
  int row = mtile * 16 + (lane & 15);
  int col = ntile * 16 + (lane & 15);
  int kd = (lane >> 4) << 1;
  const float* vp = v + (size_t)row * NUPD;
  v8f acc = {};
#pragma unroll 4
  for (int k = 0; k < NUPD; k += 4) {
    int kk = k + kd;
    v2f a; a.x = vp[kk]; a.y = vp[kk + 1];
    v2f b; b.x = dnw[(size_t)kk * DMODEL + col]; b.y = dnw[(size_t)(kk + 1) * DMODEL + col];
    acc = wmma4(a, b, acc);
  }
  float bias = dnb[col];
  int mbase = (lane >> 4) << 3;
#pragma unroll
  for (int r = 0; r < 8; ++r) {
    int orow = mtile * 16 + mbase + r;
    h[(size_t)orow * DMODEL + col] += acc[r] + bias;
  }
}

// ---------------- final projection, split-K over the 63 patches ----------------
// Partials stored [n][p][q] so both the store and the fixed-order reduction
// walk contiguous memory.
__global__ void proj_partial_kernel(const float* __restrict__ aln,
                                    const float* __restrict__ Wp,
                                    float* __restrict__ Pbuf) {
  int wave = (blockIdx.x * blockDim.x + threadIdx.x) >> 5;
  int lane = threadIdx.x & 31;
  int chunk = wave / (64 * 6);        // patch p = K-chunk
  int rem   = wave % (64 * 6);
  int mtile = rem / 6, ntile = rem % 6;
  if (chunk >= SEQ) return;
  int n = mtile * 16 + (lane & 15);
  int col = ntile * 16 + (lane & 15);
  int kd = (lane >> 4) << 1;
  const float* ap = aln + ((size_t)n * SEQ + chunk) * DMODEL;
  const float* wp = Wp + (size_t)chunk * DMODEL * PRED;
  v8f acc = {};
#pragma unroll 4
  for (int k = 0; k < DMODEL; k += 4) {
    int kk = k + kd;
    v2f a; a.x = ap[kk]; a.y = ap[kk + 1];
    v2f b; b.x = wp[(size_t)kk * PRED + col]; b.y = wp[(size_t)(kk + 1) * PRED + col];
    acc = wmma4(a, b, acc);
  }
  int mbase = (lane >> 4) << 3;
#pragma unroll
  for (int r = 0; r < 8; ++r) {
    int nn = mtile * 16 + mbase + r;
    Pbuf[((size_t)nn * SEQ + chunk) * PRED + col] = acc[r];
  }
}

// fixed-order reduction (deterministic) + bias + output transpose
__global__ void proj_reduce_kernel(const float* __restrict__ Pbuf,
                                   const float* __restrict__ bp,
                                   float* __restrict__ out) {
  int idx = blockIdx.x * blockDim.x + threadIdx.x;
  if (idx >= NSEQ * PRED) return;
  int n = idx / PRED, q = idx % PRED;
  const float* pp = Pbuf + (size_t)n * SEQ * PRED + q;
  float sum = bp[q];
  for (int p = 0; p < SEQ; ++p)
    sum += pp[(size_t)p * PRED];
  int b = n >> 5, c = n & 31;
  out[((size_t)b * PRED + q) * 32 + c] = sum;
}

extern "C" void kernel_launch(void* const* d_in, const int* in_sizes, int n_in,
                              void* d_out, int out_size, void* d_ws, size_t ws_size,
                              hipStream_t stream) {
  const float* x      = (const float*)d_in[0];
  const float* W_emb  = (const float*)d_in[1];
  const float* b_emb  = (const float*)d_in[2];
  const float* ln1_w  = (const float*)d_in[3];
  const float* ln1_b  = (const float*)d_in[4];
  const float* conv_w = (const float*)d_in[5];
  const float* conv_b = (const float*)d_in[6];
  const float* Wg     = (const float*)d_in[7];
  const float* Rg     = (const float*)d_in[8];
  const float* gb     = (const float*)d_in[9];
  const float* gn_w   = (const float*)d_in[10];
  const float* ln2_w  = (const float*)d_in[11];
  const float* ln2_b  = (const float*)d_in[12];
  const float* up_w   = (const float*)d_in[13];
  const float* up_b   = (const float*)d_in[14];
  const float* dn_w   = (const float*)d_in[15];
  const float* dn_b   = (const float*)d_in[16];
  const float* post_w = (const float*)d_in[17];
  const float* post_b = (const float*)d_in[18];
  const float* W_proj = (const float*)d_in[19];
  const float* b_proj = (const float*)d_in[20];
  float* out = (float*)d_out;

  // workspace carve-up (floats)
  float* ws = (float*)d_ws;
  size_t o = 0;
  float* h    = ws + o; o += (size_t)NROWS * DMODEL;    // residual stream
  float* Abuf = ws + o; o += (size_t)NROWS * DMODEL;    // ln outputs
  float* Bbuf = ws + o; o += (size_t)NROWS * NUPD;      // conv-silu xc / gated mlp v
  float* Ybuf = ws + o; o += (size_t)NROWS * DMODEL;    // scan outputs
  float* Pbuf = ws + o; o += (size_t)NSEQ * SEQ * PRED; // split-K partials [n][p][q]

  const int TB = 256;
  embed_kernel<<<(NROWS / 16) * 16 / 8, TB, 0, stream>>>(x, W_emb, b_emb, h);

  for (int blk = 0; blk < 2; ++blk) {
    const float* Wg_b = Wg + (size_t)blk * 4 * NHEAD * DHEAD * DHEAD;
    const float* Rg_b = Rg + (size_t)blk * 4 * NHEAD * DHEAD * DHEAD;

    ln_kernel<<<(NROWS * 32 + TB - 1) / TB, TB, 0, stream>>>(
        h, ln1_w + blk * DMODEL, ln1_b + blk * DMODEL, Abuf, NROWS);
    conv_silu_kernel<<<(int)(((size_t)NROWS * DMODEL + TB - 1) / TB), TB, 0, stream>>>(
        Abuf, conv_w + blk * 4 * DMODEL, conv_b + blk * DMODEL, Bbuf);
    // persistent scan: 64 workgroups x 16 waves, whole 63-step recurrence
    scan_kernel<<<NSEQ / 16, 512, 0, stream>>>(
        Bbuf, Abuf, Wg_b, Rg_b, gb + blk * 4 * DMODEL, Ybuf);
    gnorm_res_kernel<<<(NROWS * NHEAD * 32) / TB, TB, 0, stream>>>(
        Ybuf, gn_w + blk * DMODEL, h);
    ln_kernel<<<(NROWS * 32 + TB - 1) / TB, TB, 0, stream>>>(
        h, ln2_w + blk * DMODEL, ln2_b + blk * DMODEL, Abuf, NROWS);
    up_gelu_kernel<<<(NROWS / 16) * 24 / 8, TB, 0, stream>>>(
        Abuf, up_w + (size_t)blk * DMODEL * 2 * NUPD, up_b + blk * 2 * NUPD, Bbuf);
    down_res_kernel<<<(NROWS / 16) * 16 / 8, TB, 0, stream>>>(
        Bbuf, dn_w + (size_t)blk * NUPD * DMODEL, dn_b + blk * DMODEL, h);
  }

  ln_kernel<<<(NROWS * 32 + TB - 1) / TB, TB, 0, stream>>>(h, post_w, post_b, Abuf, NROWS);
  // split-K projection: 63 chunks x 64 row-tiles x 6 col-tiles waves
  proj_partial_kernel<<<SEQ * 64 * 6 / 8, TB, 0, stream>>>(Abuf, W_proj, Pbuf);
  proj_reduce_kernel<<<(NSEQ * PRED + TB - 1) / TB, TB, 0, stream>>>(Pbuf, b_proj, out);
}